// RelationAggregation_86354612453535
// MI455X (gfx1250) — compile-verified
//
#include <hip/hip_runtime.h>
#include <math.h>

#define N_NODES   50000
#define N_EDGES   500000
#define F         128
#define C         2
#define EDIM      64
#define NEG_SLOPE 0.2f

typedef __attribute__((ext_vector_type(2))) float v2f;
typedef __attribute__((ext_vector_type(8))) float v8f;

// Order-preserving float<->uint map so segment-max can use integer atomicMax.
__device__ __forceinline__ unsigned fenc(float f) {
    unsigned u = __float_as_uint(f);
    return (u & 0x80000000u) ? ~u : (u | 0x80000000u);
}
__device__ __forceinline__ float fdec(unsigned u) {
    return (u & 0x80000000u) ? __uint_as_float(u & 0x7FFFFFFFu) : __uint_as_float(~u);
}

// ---------------------------------------------------------------------------
// Kernel 1: x_base = x @ W_l^T   [N,128] = [N,128]x[128,128]
// One 16x16 output tile per wave via V_WMMA_F32_16X16X4_F32, K-loop of 32.
// grid = N/16 (exact), block = 256 (8 waves = 8 feature tiles).
// ---------------------------------------------------------------------------
__global__ void xbase_kernel(const float* __restrict__ x,
                             const float* __restrict__ Wl,
                             float* __restrict__ xb) {
    const int lane = threadIdx.x & 31;
    const int wave = threadIdx.x >> 5;
    const int g = lane >> 4;        // K msb group
    const int r = lane & 15;        // M (for A) / N (for B,C)
    const int m0 = blockIdx.x * 16;
    const int o0 = wave * 16;

    const float* arow = x  + (size_t)(m0 + r) * F;   // A[M=m0+r][k]
    const float* brow = Wl + (size_t)(o0 + r) * F;   // B[k][N=o0+r] = Wl[o][k]
    v8f acc = {};
    for (int k0 = 0; k0 < F; k0 += 4) {
        v2f a, b;
        a.x = arow[k0 + 2 * g];     a.y = arow[k0 + 2 * g + 1];
        b.x = brow[k0 + 2 * g];     b.y = brow[k0 + 2 * g + 1];
        acc = __builtin_amdgcn_wmma_f32_16x16x4_f32(false, a, false, b,
                                                    (short)0, acc, false, false);
    }
#pragma unroll
    for (int v = 0; v < 8; ++v)
        xb[(size_t)(m0 + v + 8 * g) * F + o0 + r] = acc[v];
}

// ---------------------------------------------------------------------------
// Shared helper: WMMA-project one 16-edge tile of edge_attr into LDS ep[16][F].
// ---------------------------------------------------------------------------
#define EP_STRIDE (F + 4)

__device__ __forceinline__ void edge_proj_tile(const float* __restrict__ ea,
                                               const float* __restrict__ We,
                                               int e0, float (*ep)[EP_STRIDE]) {
    const int lane = threadIdx.x & 31;
    const int wave = threadIdx.x >> 5;
    const int g = lane >> 4;
    const int r = lane & 15;
    const int o0 = wave * 16;

    const float* arow = ea + (size_t)(e0 + r) * EDIM;
    const float* brow = We + (size_t)(o0 + r) * EDIM;
    if (e0 + 16 < N_EDGES)  // speculative prefetch of next edge tile -> global_prefetch_b8
        __builtin_prefetch(ea + (size_t)(e0 + 16 + r) * EDIM, 0, 1);

    v8f acc = {};
    for (int k0 = 0; k0 < EDIM; k0 += 4) {
        v2f a, b;
        a.x = arow[k0 + 2 * g];     a.y = arow[k0 + 2 * g + 1];
        b.x = brow[k0 + 2 * g];     b.y = brow[k0 + 2 * g + 1];
        acc = __builtin_amdgcn_wmma_f32_16x16x4_f32(false, a, false, b,
                                                    (short)0, acc, false, false);
    }
#pragma unroll
    for (int v = 0; v < 8; ++v)
        ep[v + 8 * g][o0 + r] = acc[v];
}

// ---------------------------------------------------------------------------
// Kernel 2: per-edge attention logits + segment max (encoded atomicMax).
// Block = 16 edges; phase 1 WMMA -> LDS, phase 2: 16 threads per edge.
// ---------------------------------------------------------------------------
__global__ void edge_logits_kernel(const float* __restrict__ ea,
                                   const float* __restrict__ We,
                                   const int*   __restrict__ eidx,
                                   const float* __restrict__ xb,
                                   const float* __restrict__ cbias,   // [C,F]
                                   const float* __restrict__ ash,     // [F]
                                   const float* __restrict__ ascale,  // [C]
                                   const float* __restrict__ ebias,   // [F,C]
                                   float*       __restrict__ logits,  // [E,C]
                                   unsigned*    __restrict__ menc) {  // [N,C]
    __shared__ float ep[16][EP_STRIDE];
    const int e0 = blockIdx.x * 16;
    edge_proj_tile(ea, We, e0, ep);
    __syncthreads();

    const int el  = threadIdx.x >> 4;
    const int sub = threadIdx.x & 15;
    const int e   = e0 + el;
    const int s   = eidx[e];
    const int t   = eidx[N_EDGES + e];
    const float* xs = xb + (size_t)s * F;
    const float* xt = xb + (size_t)t * F;

    float a0 = 0.f, a1 = 0.f;
#pragma unroll
    for (int fi = 0; fi < 8; ++fi) {
        const int f = sub + fi * 16;
        const float sv = xs[f] + xt[f] + ep[el][f];
        const float w  = ash[f];
        float v0 = sv + cbias[f]     + ebias[f * 2];
        float v1 = sv + cbias[F + f] + ebias[f * 2 + 1];
        v0 = v0 > 0.f ? v0 : NEG_SLOPE * v0;
        v1 = v1 > 0.f ? v1 : NEG_SLOPE * v1;
        a0 += v0 * w;
        a1 += v1 * w;
    }
#pragma unroll
    for (int off = 8; off >= 1; off >>= 1) {
        a0 += __shfl_xor(a0, off, 16);
        a1 += __shfl_xor(a1, off, 16);
    }
    if (sub == 0) {
        const float l0 = a0 * ascale[0];
        const float l1 = a1 * ascale[1];
        logits[(size_t)e * 2]     = l0;
        logits[(size_t)e * 2 + 1] = l1;
        atomicMax(&menc[(size_t)t * 2],     fenc(l0));
        atomicMax(&menc[(size_t)t * 2 + 1], fenc(l1));
    }
}

// ---------------------------------------------------------------------------
// Kernel 3: e = exp(logit - m[tgt]); segment sums via atomicAdd.
// ---------------------------------------------------------------------------
__global__ void exp_sum_kernel(const float*    __restrict__ logits,
                               const int*      __restrict__ eidx,
                               const unsigned* __restrict__ menc,
                               float*          __restrict__ ews,     // [E,C]
                               float*          __restrict__ ssum) {  // [N,C]
    const int idx = blockIdx.x * blockDim.x + threadIdx.x;
    if (idx >= N_EDGES * C) return;
    const int e = idx >> 1;
    const int c = idx & 1;
    const int t = eidx[N_EDGES + e];
    const float m  = fdec(menc[(size_t)t * 2 + c]);
    const float ev = __expf(logits[idx] - m);
    ews[idx] = ev;
    atomicAdd(&ssum[(size_t)t * 2 + c], ev);
}

// ---------------------------------------------------------------------------
// Kernel 4: alpha + weighted-message scatter-add (recomputes edge_proj).
// out layout: out[n, c*F + f]  (channel-major concat).
// ---------------------------------------------------------------------------
__global__ void aggregate_kernel(const float* __restrict__ ea,
                                 const float* __restrict__ We,
                                 const int*   __restrict__ eidx,
                                 const float* __restrict__ xb,
                                 const float* __restrict__ ebias,
                                 const float* __restrict__ ews,
                                 const float* __restrict__ ssum,
                                 float*       __restrict__ out,
                                 float*       __restrict__ alpha) {
    __shared__ float ep[16][EP_STRIDE];
    const int e0 = blockIdx.x * 16;
    edge_proj_tile(ea, We, e0, ep);
    __syncthreads();

    const int el  = threadIdx.x >> 4;
    const int sub = threadIdx.x & 15;
    const int e   = e0 + el;
    const int s   = eidx[e];
    const int t   = eidx[N_EDGES + e];

    const float a0 = ews[(size_t)e * 2]     / (ssum[(size_t)t * 2]     + 1e-16f);
    const float a1 = ews[(size_t)e * 2 + 1] / (ssum[(size_t)t * 2 + 1] + 1e-16f);
    if (sub == 0) {
        alpha[(size_t)e * 2]     = a0;
        alpha[(size_t)e * 2 + 1] = a1;
    }

    const float* xs   = xb + (size_t)s * F;
    float*       orow = out + (size_t)t * (C * F);
#pragma unroll
    for (int fi = 0; fi < 8; ++fi) {
        const int f = sub + fi * 16;
        const float base = xs[f] + ep[el][f];
        atomicAdd(&orow[f],     (base + ebias[f * 2])     * a0);
        atomicAdd(&orow[F + f], (base + ebias[f * 2 + 1]) * a1);
    }
}

// ---------------------------------------------------------------------------
extern "C" void kernel_launch(void* const* d_in, const int* in_sizes, int n_in,
                              void* d_out, int out_size, void* d_ws, size_t ws_size,
                              hipStream_t stream) {
    (void)in_sizes; (void)n_in; (void)out_size; (void)ws_size;

    const float* x     = (const float*)d_in[0];
    const int*   eidx  = (const int*)  d_in[1];
    const float* ea    = (const float*)d_in[2];
    const float* Wl    = (const float*)d_in[3];
    const float* cbias = (const float*)d_in[4];
    const float* ash   = (const float*)d_in[5];
    const float* ascal = (const float*)d_in[6];
    const float* We    = (const float*)d_in[7];
    const float* ebias = (const float*)d_in[8];

    float* out   = (float*)d_out;                       // [N, C*F]
    float* alpha = out + (size_t)N_NODES * C * F;       // [E, C]

    char* w = (char*)d_ws;
    unsigned* menc   = (unsigned*)w;  w += sizeof(unsigned) * (size_t)N_NODES * C;
    float*    ssum   = (float*)w;     w += sizeof(float)    * (size_t)N_NODES * C;
    float*    logits = (float*)w;     w += sizeof(float)    * (size_t)N_EDGES * C;
    float*    ews    = (float*)w;     w += sizeof(float)    * (size_t)N_EDGES * C;
    float*    xb     = (float*)w;     w += sizeof(float)    * (size_t)N_NODES * F;

    hipMemsetAsync(menc, 0, sizeof(unsigned) * (size_t)N_NODES * C, stream);
    hipMemsetAsync(ssum, 0, sizeof(float)    * (size_t)N_NODES * C, stream);
    hipMemsetAsync(out,  0, sizeof(float)    * (size_t)N_NODES * C * F, stream);

    xbase_kernel      <<<N_NODES / 16, 256, 0, stream>>>(x, Wl, xb);
    edge_logits_kernel<<<N_EDGES / 16, 256, 0, stream>>>(ea, We, eidx, xb, cbias,
                                                         ash, ascal, ebias, logits, menc);
    const int tot = N_EDGES * C;
    exp_sum_kernel    <<<(tot + 255) / 256, 256, 0, stream>>>(logits, eidx, menc, ews, ssum);
    aggregate_kernel  <<<N_EDGES / 16, 256, 0, stream>>>(ea, We, eidx, xb, ebias,
                                                         ews, ssum, out, alpha);
}